// StackLSTM_13460427506023
// MI455X (gfx1250) — compile-verified
//
#include <hip/hip_runtime.h>
#include <math.h>

typedef float    v2f __attribute__((ext_vector_type(2)));
typedef float    v8f __attribute__((ext_vector_type(8)));
typedef unsigned v4u __attribute__((ext_vector_type(4)));
typedef int      v4i __attribute__((ext_vector_type(4)));
typedef int      v8i __attribute__((ext_vector_type(8)));

#define TT   256
#define BB   32
#define DD0  512
#define HH   1024
#define NWG  64               // one WG per 16 hidden units
#define NTHR 256
#define KTOT0 (DD0 + HH)      // 1536
#define KTOT1 (HH + HH)       // 2048
#define KC0  (KTOT0 / 4)      // 384
#define KC1  (KTOT1 / 4)      // 512
#define LPITCH 132            // 132 % 64 == 4 -> conflict-free column reads
#define LDSSZ (32 * LPITCH)   // floats per activation slot (4224)

__device__ __forceinline__ float sigmoidf_(float x) { return 1.0f / (1.0f + __expf(-x)); }

// ---------------------------------------------------------------------------
// Device-wide barrier: monotonic counter, release/acquire via threadfence.
// ---------------------------------------------------------------------------
__device__ __forceinline__ void grid_barrier(int* cnt, int target) {
    __threadfence();
    __syncthreads();
    if (threadIdx.x == 0) {
        __hip_atomic_fetch_add(cnt, 1, __ATOMIC_RELAXED, __HIP_MEMORY_SCOPE_AGENT);
        while (__hip_atomic_load(cnt, __ATOMIC_RELAXED, __HIP_MEMORY_SCOPE_AGENT) < target) {
            __builtin_amdgcn_s_sleep(2);
        }
    }
    __syncthreads();
    __threadfence();
}

// ---------------------------------------------------------------------------
// TDM: DMA a 32-row x 128-float tile (row stride pitchElems) from global into
// LDS at ldsByteAddr, with HW padding of +4 DWORDs every 128 DWORDs so the
// landed LDS pitch is 132 floats (bank-conflict-free column reads).
// D# fields per CDNA5 ISA ch.8 (group0 128b, group1 256b; higher groups zero).
// ---------------------------------------------------------------------------
__device__ __forceinline__ void tdm_issue(const float* src, int pitchElems,
                                          unsigned ldsByteAddr)
{
    unsigned long long ga = (unsigned long long)(uintptr_t)src;
    v4u g0; v8i g1;
    v4i gz4 = {0, 0, 0, 0};
    v8i gz8 = {0, 0, 0, 0, 0, 0, 0, 0};
    g0[0] = 1u;                                        // count=1, user D#
    g0[1] = ldsByteAddr;                               // lds_addr
    g0[2] = (unsigned)(ga & 0xFFFFFFFFu);              // global_addr[31:0]
    g0[3] = (unsigned)((ga >> 32) & 0x1FFFFFFu)        // global_addr[56:32]
          | (2u << 30);                                // type = 2 (image)
    g1[0] = (int)((2u << 16)                           // data_size = 4B
          | (1u << 20)                                 // pad_enable
          | (6u << 22)                                 // pad_interval: 128 DW
          | (3u << 25));                               // pad_amount:   4 DW
    g1[1] = (int)(128u << 16);                         // tensor_dim0 = 128
    g1[2] = (int)(32u << 16);                          // tensor_dim1 = 32
    g1[3] = (int)(128u << 16);                         // tile_dim0   = 128
    g1[4] = 32;                                        // tile_dim1   = 32
    g1[5] = pitchElems;                                // tensor_dim0_stride
    g1[6] = 0;
    g1[7] = 0;
    __builtin_amdgcn_tensor_load_to_lds(g0, g1, gz4, gz4, gz8, 0);
}

// Issue the chunk-pair (both K-halves) for iteration `it` into LDS slot it%3.
__device__ __forceinline__ void issue_pair(
    const float* __restrict__ src0, int pitch0, int d0,
    const float* __restrict__ src1, int pitch1,
    int khalf, int it, unsigned lds_base_b)
{
    int kA = it * 128;
    int kB = khalf + it * 128;
    const float* sA = (kA < d0) ? (src0 + kA) : (src1 + (kA - d0));
    int          pA = (kA < d0) ? pitch0 : pitch1;
    const float* sB = (kB < d0) ? (src0 + kB) : (src1 + (kB - d0));
    int          pB = (kB < d0) ? pitch0 : pitch1;
    unsigned slot = (unsigned)(it % 3) * 2u;
    tdm_issue(sA, pA, lds_base_b + (slot + 0u) * (LDSSZ * 4u));
    tdm_issue(sB, pB, lds_base_b + (slot + 1u) * (LDSSZ * 4u));
}

// ---------------------------------------------------------------------------
// K-split gate GEMM for one wave: D[16x32] partial (acc0/acc1) over K-half `ks`.
// TDM stages chunk it+1 while all waves WMMA over chunk it; 3-slot rotation
// keeps exactly one __syncthreads per chunk.
// ---------------------------------------------------------------------------
__device__ __forceinline__ void gates_matmul(
    const float* __restrict__ src0, int pitch0, int d0,
    const float* __restrict__ src1, int pitch1, int ktot,
    const float* __restrict__ apack_wave,   // + (mtile*KC + ks*KC/2)*64
    float* lds, unsigned lds_base_b, int ks, int wv,
    v8f& acc0, v8f& acc1)
{
    const int lane  = threadIdx.x & 31;
    const int dk    = (lane >> 4) * 2;      // K sub-offset per ISA f32 A/B layout
    const int n0    = lane & 15;
    const int khalf = ktot >> 1;
    const int nIter = khalf >> 7;           // 128-K chunks per half
    const float* ap = apack_wave + lane * 2;

    acc0 = (v8f){0.f,0.f,0.f,0.f,0.f,0.f,0.f,0.f};
    acc1 = (v8f){0.f,0.f,0.f,0.f,0.f,0.f,0.f,0.f};

    if (wv == 0) issue_pair(src0, pitch0, d0, src1, pitch1, khalf, 0, lds_base_b);

    for (int it = 0; it < nIter; ++it) {
        if (wv == 0) {
            if (it + 1 < nIter) {
                issue_pair(src0, pitch0, d0, src1, pitch1, khalf, it + 1, lds_base_b);
                __builtin_amdgcn_s_wait_tensorcnt(2);   // chunk `it` landed
            } else {
                __builtin_amdgcn_s_wait_tensorcnt(0);   // final chunk landed
            }
        }
        __syncthreads();

        __builtin_prefetch((const void*)(ap + 32 * 64), 0, 1);   // next A block

        const float* bb = lds + (size_t)((it % 3) * 2 + ks) * LDSSZ;
        #pragma unroll 8
        for (int kc = 0; kc < 128; kc += 4) {
            v2f a = *(const v2f*)ap;        // A frag: W[m][dk], W[m][dk+1]
            ap += 64;
            int kb = kc + dk;               // even -> 8B-aligned LDS b64
            v2f b0 = *(const v2f*)(bb + n0 * LPITCH + kb);
            v2f b1 = *(const v2f*)(bb + (n0 + 16) * LPITCH + kb);
            acc0 = __builtin_amdgcn_wmma_f32_16x16x4_f32(
                false, a, false, b0, (short)0, acc0, false, false);
            acc1 = __builtin_amdgcn_wmma_f32_16x16x4_f32(
                false, a, false, b1, (short)0, acc1, false, false);
        }
        // no trailing sync: TDM writes slot (it+1)%3, compute reads it%3, and
        // slot (it+2)%3 was fully consumed before the previous barrier.
    }
}

// Scatter one 16x16 partial tile into LDS: gbuf[ks][g][u(16)][n(33 pitched)]
__device__ __forceinline__ void store_acc(
    float* gb, int ks, int g, int lane, v8f acc, int nt)
{
    int n  = (lane & 15) + nt * 16;
    int mo = (lane >> 4) * 8;               // C/D: lanes16-31 hold M=r+8
    float* p = gb + (size_t)((ks * 4 + g) * 16) * 33;
    #pragma unroll
    for (int r = 0; r < 8; ++r)
        p[(mo + r) * 33 + n] = acc[r];
}

// Fused LSTM pointwise: reduce K-split partials + gates for 16 units x 32 batch.
__device__ __forceinline__ void lstm_pointwise(
    const float* gb, const float* __restrict__ bsum,
    float* __restrict__ cbuf, float* __restrict__ hout,
    int w, float* __restrict__ out)
{
    #pragma unroll
    for (int i = 0; i < 2; ++i) {
        int idx = i * NTHR + threadIdx.x;   // 0..511
        int u   = idx & 15;                 // local unit (coalesced stores)
        int n   = idx >> 4;                 // batch
        int col = w * 16 + u;
        float xi = gb[(0*16 + u)*33 + n] + gb[((4+0)*16 + u)*33 + n] + bsum[0*HH + col];
        float xf = gb[(1*16 + u)*33 + n] + gb[((4+1)*16 + u)*33 + n] + bsum[1*HH + col];
        float xg = gb[(2*16 + u)*33 + n] + gb[((4+2)*16 + u)*33 + n] + bsum[2*HH + col];
        float xo = gb[(3*16 + u)*33 + n] + gb[((4+3)*16 + u)*33 + n] + bsum[3*HH + col];
        float iv = sigmoidf_(xi);
        float fv = sigmoidf_(xf);
        float gv = tanhf(xg);
        float ov = sigmoidf_(xo);
        float cold = cbuf[n * HH + col];
        float cnew = fv * cold + iv * gv;
        float hnew = ov * tanhf(cnew);
        cbuf[n * HH + col] = cnew;
        hout[n * HH + col] = hnew;
        if (out) out[n * HH + col] = hnew;
    }
}

// ---------------------------------------------------------------------------
// Persistent 2-layer LSTM over all 256 timesteps. 64 WGs x 8 waves.
// Wave v: gate g = v&3, K-half ks = v>>2; WG w owns hidden units [w*16, w*16+16).
// ---------------------------------------------------------------------------
__global__ __launch_bounds__(NTHR, 1) void lstm_persist(
    const float* __restrict__ xseq,
    const float* __restrict__ bsum0, const float* __restrict__ bsum1,
    const float* __restrict__ packed0, const float* __restrict__ packed1,
    float* h0b, float* h1b, float* c0b, float* c1b,
    int* bar, float* out)
{
    __shared__ __align__(16) float bufs[6 * LDSSZ];       // 3 slots x 2 K-halves, 101 KB
    __shared__ __align__(16) float gbuf[2 * 4 * 16 * 33]; // partial gates, 16.9 KB

    const int tid  = threadIdx.x;
    const int lane = tid & 31;
    const int wv   = tid >> 5;
    const int w    = blockIdx.x;
    const int g    = wv & 3;
    const int ks   = wv >> 2;
    const int mtile = g * 64 + w;
    const unsigned bufs_lds = (unsigned)(uintptr_t)(void*)bufs;  // LDS byte offset

    const float* ap0 = packed0 + ((size_t)mtile * KC0 + (size_t)ks * (KC0 / 2)) * 64;
    const float* ap1 = packed1 + ((size_t)mtile * KC1 + (size_t)ks * (KC1 / 2)) * 64;

    int barphase = 0;
    for (int t = 0; t < TT; ++t) {
        const int cur = t & 1, nxt = cur ^ 1;
        v8f a0, a1;

        // ----- layer 0 -----
        gates_matmul(xseq + (size_t)t * BB * DD0, DD0, DD0,
                     h0b + (size_t)cur * BB * HH, HH, KTOT0,
                     ap0, bufs, bufs_lds, ks, wv, a0, a1);
        store_acc(gbuf, ks, g, lane, a0, 0);
        store_acc(gbuf, ks, g, lane, a1, 1);
        __syncthreads();
        lstm_pointwise(gbuf, bsum0, c0b, h0b + (size_t)nxt * BB * HH, w, nullptr);
        grid_barrier(bar, NWG * (++barphase));

        // ----- layer 1 -----
        gates_matmul(h0b + (size_t)nxt * BB * HH, HH, HH,
                     h1b + (size_t)cur * BB * HH, HH, KTOT1,
                     ap1, bufs, bufs_lds, ks, wv, a0, a1);
        store_acc(gbuf, ks, g, lane, a0, 0);
        store_acc(gbuf, ks, g, lane, a1, 1);
        __syncthreads();
        lstm_pointwise(gbuf, bsum1, c1b, h1b + (size_t)nxt * BB * HH, w,
                       (t == TT - 1) ? out : nullptr);
        grid_barrier(bar, NWG * (++barphase));
    }
}

// ---------------------------------------------------------------------------
// Weight repack into WMMA A-fragment order (K = [W_ih | W_hh] fused):
// packed[((mt*KC + kc)*64) + lane*2 + r] = W[mt*16 + (lane&15)][kc*4 + (lane>>4)*2 + r]
// ---------------------------------------------------------------------------
__global__ void repack_w(const float* __restrict__ Wih,
                         const float* __restrict__ Whh,
                         int Din, int ktot, float* __restrict__ dst)
{
    size_t p = (size_t)blockIdx.x * blockDim.x + threadIdx.x;
    size_t total = (size_t)4 * HH * ktot;
    if (p >= total) return;
    int KCq = ktot >> 2;
    int r    = (int)(p & 1);
    int lane = (int)((p >> 1) & 31);
    size_t q = p >> 6;
    int kc = (int)(q % KCq);
    int mt = (int)(q / KCq);
    int row = mt * 16 + (lane & 15);
    int k   = kc * 4 + (lane >> 4) * 2 + r;
    float v = (k < Din) ? Wih[(size_t)row * Din + k]
                        : Whh[(size_t)row * HH + (k - Din)];
    dst[p] = v;
}

__global__ void init_state(const float* __restrict__ h0in, const float* __restrict__ c0in,
                           const float* __restrict__ bih0, const float* __restrict__ bhh0,
                           const float* __restrict__ bih1, const float* __restrict__ bhh1,
                           float* h0b, float* h1b, float* c0b, float* c1b,
                           float* bsum0, float* bsum1, int* bar)
{
    int i = blockIdx.x * blockDim.x + threadIdx.x;
    if (i < BB * HH) {
        h0b[i] = h0in[i];             // layer0 h  (h0[0])
        h1b[i] = h0in[BB * HH + i];   // layer1 h  (h0[1])
        c0b[i] = c0in[i];
        c1b[i] = c0in[BB * HH + i];
    }
    if (i < 4 * HH) {
        bsum0[i] = bih0[i] + bhh0[i];
        bsum1[i] = bih1[i] + bhh1[i];
    }
    if (i == 0) *bar = 0;
}

extern "C" void kernel_launch(void* const* d_in, const int* in_sizes, int n_in,
                              void* d_out, int out_size, void* d_ws, size_t ws_size,
                              hipStream_t stream) {
    const float* xseq = (const float*)d_in[0];
    const float* h0   = (const float*)d_in[1];
    const float* c0   = (const float*)d_in[2];
    const float* Wih0 = (const float*)d_in[3];
    const float* Whh0 = (const float*)d_in[4];
    const float* bih0 = (const float*)d_in[5];
    const float* bhh0 = (const float*)d_in[6];
    const float* Wih1 = (const float*)d_in[7];
    const float* Whh1 = (const float*)d_in[8];
    const float* bih1 = (const float*)d_in[9];
    const float* bhh1 = (const float*)d_in[10];

    float* ws      = (float*)d_ws;
    float* packed0 = ws;                                   // 4096*1536 floats
    float* packed1 = packed0 + (size_t)4 * HH * KTOT0;     // 4096*2048 floats
    float* h0b     = packed1 + (size_t)4 * HH * KTOT1;     // 2 * 32*1024
    float* h1b     = h0b + 2 * BB * HH;
    float* c0b     = h1b + 2 * BB * HH;
    float* c1b     = c0b + BB * HH;
    float* bsum0   = c1b + BB * HH;                        // 4*1024
    float* bsum1   = bsum0 + 4 * HH;
    int*   bar     = (int*)(bsum1 + 4 * HH);

    size_t n0 = (size_t)4 * HH * KTOT0;
    size_t n1 = (size_t)4 * HH * KTOT1;
    repack_w<<<(unsigned)((n0 + 255) / 256), 256, 0, stream>>>(Wih0, Whh0, DD0, KTOT0, packed0);
    repack_w<<<(unsigned)((n1 + 255) / 256), 256, 0, stream>>>(Wih1, Whh1, HH,  KTOT1, packed1);
    init_state<<<(BB * HH + 255) / 256, 256, 0, stream>>>(
        h0, c0, bih0, bhh0, bih1, bhh1, h0b, h1b, c0b, c1b, bsum0, bsum1, bar);

    lstm_persist<<<NWG, NTHR, 0, stream>>>(
        xseq, bsum0, bsum1, packed0, packed1,
        h0b, h1b, c0b, c1b, bar, (float*)d_out);
}